// EdgeConvBlock_16381005267563
// MI455X (gfx1250) — compile-verified
//
// EdgeConv (DGCNN) block for MI455X / gfx1250.
//
// Memory-bound workload (39 GFLOP vs ~2.5 GB HBM traffic -> ~110us @ 23.3TB/s),
// so we keep full fp32 precision and use V_WMMA_F32_16X16X4_F32 chained 16x
// along K=64. Each wave32 owns a 16-edge x 64-feature tile.
//
// Phases (BatchNorm batch statistics force sequential passes):
//   P1: m=[x_i, x_j-x_i] -> a1 = m@W1^T+b1 (WMMA), store [E,64], accum sum/sq
//   F1: fold mu/var/gamma/beta into per-feature scale/shift
//   P2: h1=relu(bn(a1)) -> a2 (WMMA), in-place, accum stats ; F2
//   P3: h2 -> a3 (WMMA), in-place, accum stats ; F3
//   P4: h3=relu(bn(a3)) scatter-add into out[dst] via global_atomic_add_f32
//
// Workspace: 4KB stats header + E*64*4 B (~410 MB) activation buffer.

#include <hip/hip_runtime.h>
#include <math.h>

typedef __attribute__((ext_vector_type(2))) float v2f;
typedef __attribute__((ext_vector_type(8))) float v8f;

#define WAVES_PER_BLOCK 8
#define EDGES_PER_WAVE  16
#define EDGES_PER_BLOCK 128
#define H_DIM           64
#define ROW_STRIDE      68   // padded LDS row stride (floats): banks = 4*row+k, conflict-free

__device__ __forceinline__ float relu_bn(float a, float sc, float sh) {
  float v = fmaf(a, sc, sh);
  return v > 0.f ? v : 0.f;
}

// MODE 0: stage m = [x_i, x_j - x_i] from gathered X rows.
// MODE 1: stage h = relu(bn(a_prev)) from the activation buffer.
template<int MODE>
__global__ __launch_bounds__(256)
void edgeconv_gemm_pass(const float* __restrict__ X,
                        const int* __restrict__ srcI,
                        const int* __restrict__ dstI,
                        const float* __restrict__ Ain,
                        const float* __restrict__ bnScale,
                        const float* __restrict__ bnShift,
                        const float* __restrict__ W,      // [64 out][64 in] row-major
                        const float* __restrict__ bias,   // [64]
                        float* __restrict__ Aout,         // [E,64] pre-activations
                        float* __restrict__ statSum,      // [64]
                        float* __restrict__ statSq,       // [64]
                        int E)
{
  __shared__ float sW[H_DIM * ROW_STRIDE];                           // 17.4 KB
  __shared__ float sM[WAVES_PER_BLOCK][EDGES_PER_WAVE * ROW_STRIDE]; // 34.8 KB
  __shared__ float sSum[H_DIM];
  __shared__ float sSq[H_DIM];

  const int tid  = threadIdx.x;
  const int wave = tid >> 5;
  const int lane = tid & 31;
  const int m    = lane & 15;   // edge-in-tile (matrix row M)
  const int half = lane >> 4;   // lane half selects K sub-pair / M offset
  const int eBase = (blockIdx.x * WAVES_PER_BLOCK + wave) * EDGES_PER_WAVE;
  const bool fullTile = (eBase + EDGES_PER_WAVE) <= E;   // wave-uniform

  // ---- stage weights: sW[n][k] = W[n][k]  (B = W^T, so lane reads contiguous k-pairs)
  for (int i = tid; i < H_DIM * 16; i += 256) {
    int n = i >> 4, q = i & 15;
    float4 w = ((const float4*)W)[n * 16 + q];
    *(float4*)&sW[n * ROW_STRIDE + q * 4] = w;
  }
  if (tid < H_DIM) { sSum[tid] = 0.f; sSq[tid] = 0.f; }

  // ---- stage this wave's 16x64 activation tile (each lane: 8 float4 = half an edge row)
  {
    int e = eBase + m;
    bool valid = (e < E);
    if (MODE == 0) {
      int ei = valid ? dstI[e] : 0;   // x_i = X[dst]
      int ej = valid ? srcI[e] : 0;   // x_j = X[src]
      const float4* xi = (const float4*)(X + (size_t)ei * 32);
      const float4* xj = (const float4*)(X + (size_t)ej * 32);
#pragma unroll
      for (int q = 0; q < 8; ++q) {
        float4 v = make_float4(0.f, 0.f, 0.f, 0.f);
        if (valid) {
          float4 a = xi[q];
          if (half == 0) {
            v = a;                                        // features 0..31 : x_i
          } else {
            float4 b = xj[q];                             // features 32..63: x_j - x_i
            v = make_float4(b.x - a.x, b.y - a.y, b.z - a.z, b.w - a.w);
          }
        }
        *(float4*)&sM[wave][m * ROW_STRIDE + half * 32 + q * 4] = v;
      }
    } else {
      const float4* ap = (const float4*)(Ain + (size_t)e * 64 + half * 32);
      // Hint the next block's tile toward L2 (global_prefetch_b8).
      __builtin_prefetch(Ain + (size_t)(e + EDGES_PER_BLOCK) * 64 + half * 32, 0, 1);
#pragma unroll
      for (int q = 0; q < 8; ++q) {
        int f = half * 32 + q * 4;
        float4 v = make_float4(0.f, 0.f, 0.f, 0.f);
        if (valid) {
          float4 a  = ap[q];
          float4 sc = *(const float4*)(bnScale + f);
          float4 sh = *(const float4*)(bnShift + f);
          v.x = relu_bn(a.x, sc.x, sh.x);
          v.y = relu_bn(a.y, sc.y, sh.y);
          v.z = relu_bn(a.z, sc.z, sh.z);
          v.w = relu_bn(a.w, sc.w, sh.w);
        }
        *(float4*)&sM[wave][m * ROW_STRIDE + half * 32 + q * 4] = v;
      }
    }
  }
  __syncthreads();

  // ---- per-lane A fragments: A(16x4 f32) layout: lanes 0-15 K=4k+{0,1}, lanes 16-31 K=4k+{2,3}
  const float* mrow = &sM[wave][m * ROW_STRIDE];
  const int koff = half * 2;

  // ---- 4 N-tiles of 16 output features, K=64 via 16 chained 16x16x4 f32 WMMAs
#pragma unroll
  for (int nt = 0; nt < 4; ++nt) {
    const int f = nt * 16 + m;             // this lane's output column (feature)
    const float* wrow = &sW[f * ROW_STRIDE];
    v8f c = {};
#pragma unroll
    for (int ks = 0; ks < 16; ++ks) {
      v2f afrag = *(const v2f*)(mrow + ks * 4 + koff);
      v2f bfrag = *(const v2f*)(wrow + ks * 4 + koff);  // B[k][n] = W[n][k]
      c = __builtin_amdgcn_wmma_f32_16x16x4_f32(false, afrag, false, bfrag,
                                                (short)0, c, false, false);
    }
    const float bv = bias[f];
    float s = 0.f, s2 = 0.f;
    if (fullTile) {
      // Fast path: unpredicated stores from one base address + immediate offsets.
      float* op = Aout + (size_t)(eBase + half * 8) * 64 + f;
#pragma unroll
      for (int r = 0; r < 8; ++r) {        // C layout: VGPR r -> edge M = r + 8*half
        float val = c[r] + bv;
        op[(size_t)r * 64] = val;
        s  += val;
        s2 += val * val;
      }
    } else {
#pragma unroll
      for (int r = 0; r < 8; ++r) {
        int e = eBase + r + half * 8;
        if (e < E) {
          float val = c[r] + bv;
          Aout[(size_t)e * 64 + f] = val;
          s  += val;
          s2 += val * val;
        }
      }
    }
    atomicAdd(&sSum[f], s);                // ds_add_f32, 2-way conflict max
    atomicAdd(&sSq[f],  s2);
  }
  __syncthreads();

  if (tid < H_DIM) {
    atomicAdd(&statSum[tid], sSum[tid]);
    atomicAdd(&statSq[tid],  sSq[tid]);
  }
}

__global__ void edgeconv_finalize_bn(const float* __restrict__ sum,
                                     const float* __restrict__ sq,
                                     const float* __restrict__ g,
                                     const float* __restrict__ be,
                                     float* __restrict__ scale,
                                     float* __restrict__ shift,
                                     float invE)
{
  int f = threadIdx.x;           // 64 threads
  float mu  = sum[f] * invE;
  float var = fmaxf(fmaf(-mu, mu, sq[f] * invE), 0.f);
  float rs  = rsqrtf(var + 1e-5f);
  float sc  = rs * g[f];
  scale[f] = sc;
  shift[f] = fmaf(-mu, sc, be[f]);
}

__global__ __launch_bounds__(256)
void edgeconv_scatter(const float* __restrict__ A,
                      const float* __restrict__ scale,
                      const float* __restrict__ shift,
                      const int* __restrict__ dstI,
                      float* __restrict__ out,
                      int E)
{
  long long i = (long long)blockIdx.x * 256 + threadIdx.x;  // one float4 group / thread
  int e = (int)(i >> 4);
  int q = (int)(i & 15);
  if (e >= E) return;
  int f = q * 4;
  float4 a  = ((const float4*)(A + (size_t)e * 64))[q];
  float4 sc = *(const float4*)(scale + f);
  float4 sh = *(const float4*)(shift + f);
  float4 v;
  v.x = relu_bn(a.x, sc.x, sh.x);
  v.y = relu_bn(a.y, sc.y, sh.y);
  v.z = relu_bn(a.z, sc.z, sh.z);
  v.w = relu_bn(a.w, sc.w, sh.w);
  float* o = out + (size_t)dstI[e] * 64 + f;
  atomicAdd(o + 0, v.x);
  atomicAdd(o + 1, v.y);
  atomicAdd(o + 2, v.z);
  atomicAdd(o + 3, v.w);
}

extern "C" void kernel_launch(void* const* d_in, const int* in_sizes, int n_in,
                              void* d_out, int out_size, void* d_ws, size_t ws_size,
                              hipStream_t stream) {
  const float* X   = (const float*)d_in[0];
  const int*   ei  = (const int*)d_in[1];     // [2,E] flat, int32
  const float* W1  = (const float*)d_in[2];
  const float* b1  = (const float*)d_in[3];
  const float* g1  = (const float*)d_in[4];
  const float* be1 = (const float*)d_in[5];
  const float* W2  = (const float*)d_in[6];
  const float* b2  = (const float*)d_in[7];
  const float* g2  = (const float*)d_in[8];
  const float* be2 = (const float*)d_in[9];
  const float* W3  = (const float*)d_in[10];
  const float* b3  = (const float*)d_in[11];
  const float* g3  = (const float*)d_in[12];
  const float* be3 = (const float*)d_in[13];

  const int E = in_sizes[1] / 2;
  const int* srcI = ei;
  const int* dstI = ei + E;

  // ws layout (floats): [0:384) raw stats (sum/sq x3), [384:768) scale/shift x3,
  // then 4KB-aligned activation buffer [E,64].  Requires ws_size >= 4096 + E*256 B.
  float* st   = (float*)d_ws;
  float* Abuf = (float*)((char*)d_ws + 4096);

  hipMemsetAsync(d_out, 0, sizeof(float) * (size_t)out_size, stream);
  hipMemsetAsync(d_ws, 0, 4096, stream);

  const int gemmBlocks = (E + EDGES_PER_BLOCK - 1) / EDGES_PER_BLOCK;
  const float invE = 1.0f / (float)E;

  edgeconv_gemm_pass<0><<<gemmBlocks, 256, 0, stream>>>(
      X, srcI, dstI, nullptr, nullptr, nullptr,
      W1, b1, Abuf, st + 0, st + 64, E);
  edgeconv_finalize_bn<<<1, 64, 0, stream>>>(st + 0, st + 64, g1, be1,
                                             st + 384, st + 448, invE);

  edgeconv_gemm_pass<1><<<gemmBlocks, 256, 0, stream>>>(
      nullptr, srcI, dstI, Abuf, st + 384, st + 448,
      W2, b2, Abuf, st + 128, st + 192, E);
  edgeconv_finalize_bn<<<1, 64, 0, stream>>>(st + 128, st + 192, g2, be2,
                                             st + 512, st + 576, invE);

  edgeconv_gemm_pass<1><<<gemmBlocks, 256, 0, stream>>>(
      nullptr, srcI, dstI, Abuf, st + 512, st + 576,
      W3, b3, Abuf, st + 256, st + 320, E);
  edgeconv_finalize_bn<<<1, 64, 0, stream>>>(st + 256, st + 320, g3, be3,
                                             st + 640, st + 704, invE);

  long long groups = (long long)E * 16;
  int scatBlocks = (int)((groups + 255) / 256);
  edgeconv_scatter<<<scatBlocks, 256, 0, stream>>>(
      Abuf, st + 640, st + 704, dstI, (float*)d_out, E);
}